// RecurrentHGAT_49443663511613
// MI455X (gfx1250) — compile-verified
//
#include <hip/hip_runtime.h>

// ---------------------------------------------------------------------------
// RecurrentHGAT step for MI455X (gfx1250, wave32).
// Dense GEMMs: fp32 WMMA (V_WMMA_F32_16X16X4_F32), TN form, branch-free loop,
// two 16x16 C tiles per wave. A-tile staged via GLOBAL_LOAD_ASYNC_TO_LDS
// (ASYNCcnt). GAT attention dots fused into the GEMM epilogue (cross-lane
// shuffle reduction + atomicAdd) so hs/hd are never re-read.
// Edge scatter: wave-per-edge with global_prefetch of the next gathered row.
// ---------------------------------------------------------------------------

#define HEADS 4
#define CH    64
#define HC    256      // HEADS*CH
#define G3    192      // 3*CH
#define NINT  50000
#define NLANE 400000
#define EPO   400000
#define EADJ  200000

typedef float v2f __attribute__((ext_vector_type(2)));
typedef float v8f __attribute__((ext_vector_type(8)));
typedef __attribute__((address_space(3))) float lds_float;

// 16-byte int vector in global (AS1) / LDS (AS3) address spaces for the
// async-copy builtin signature.
typedef int v4i_ __attribute__((vector_size(4 * sizeof(int))));
typedef __attribute__((address_space(1))) v4i_ g_v4i;
typedef __attribute__((address_space(3))) v4i_ l_v4i;

__device__ __forceinline__ float sigm_(float x) { return 1.0f / (1.0f + __expf(-x)); }
__device__ __forceinline__ float tanh_(float x) { return 1.0f - 2.0f / (__expf(2.0f * x) + 1.0f); }

// float atomic max via ordered-int mapping (init with -inf)
__device__ __forceinline__ void atomicMaxF(float* addr, float val) {
  if (val >= 0.0f) atomicMax((int*)addr, __float_as_int(val));
  else             atomicMin((unsigned int*)addr, (unsigned int)__float_as_int(val));
}

// 16B async copy global -> LDS (GLOBAL_LOAD_ASYNC_TO_LDS_B128, ASYNCcnt)
__device__ __forceinline__ void async_copy16(const float* g, lds_float* l) {
#if __has_builtin(__builtin_amdgcn_global_load_async_to_lds_b128)
  __builtin_amdgcn_global_load_async_to_lds_b128((g_v4i*)g, (l_v4i*)l, 0, 0);
#else
  unsigned off = (unsigned)(uintptr_t)l;
  asm volatile("global_load_async_to_lds_b128 %0, %1, off" :: "v"(off), "v"(g) : "memory");
#endif
}

__device__ __forceinline__ void wait_async0() {
#if __has_builtin(__builtin_amdgcn_s_wait_asynccnt)
  __builtin_amdgcn_s_wait_asynccnt(0);
#else
  asm volatile("s_wait_asynccnt 0x0" ::: "memory");
#endif
}

// ---------------------------------------------------------------------------
__global__ void fill_kernel(float* __restrict__ p, int n, float v) {
  int t = blockIdx.x * blockDim.x + threadIdx.x;
  if (t < n) p[t] = v;
}

// WT[n,k] = W[k,n]  (W is [K][Nn] row-major)
__global__ void transpose_kernel(const float* __restrict__ W, float* __restrict__ WT,
                                 int K, int Nn) {
  int t = blockIdx.x * blockDim.x + threadIdx.x;
  if (t >= K * Nn) return;
  int k = t / Nn, n = t - k * Nn;
  WT[(size_t)n * K + k] = W[t];
}

// out[n,c] = relu(x[n,:] @ W[:,c] + b[c]),  c in [0,64)
__global__ void enc_kernel(const float* __restrict__ x, const float* __restrict__ W,
                           const float* __restrict__ b, float* __restrict__ out,
                           int Nn, int Kin) {
  int t = blockIdx.x * blockDim.x + threadIdx.x;
  if (t >= Nn * CH) return;
  int n = t >> 6, c = t & 63;
  const float* xr = x + (size_t)n * Kin;
  float acc = b[c];
  for (int k = 0; k < Kin; ++k) acc += xr[k] * W[k * CH + c];
  out[t] = fmaxf(acc, 0.0f);
}

// ---------------------------------------------------------------------------
// TN GEMM: C[M,N] = A[M,K] @ Bt[N,K]^T.  M,N multiples of 16, K multiple of 4.
// grid.x = M/16; block = 256 (8 waves). Wave w computes N-tiles 2w and 2w+1.
// Optional fused attention dot: avec[row,h] += sum_c C[row, h*64+c]*att[h*64+c]
// (valid when N == HC; a wave's 32 columns lie inside one head).
// fp32 WMMA layouts per CDNA5 ISA 7.12.2:
//   A 16x4: elem(M=lane&15, K=2*(lane>>4)+v)   v = vgpr 0/1
//   B 4x16: elem(K=2*(lane>>4)+v, N=lane&15)
//   C 16x16: elem(M=v+8*(lane>>4), N=lane&15)  v = vgpr 0..7
__global__ void wmma_gemm_tn(const float* __restrict__ A, const float* __restrict__ Bt,
                             float* __restrict__ C, int M, int N, int K,
                             const float* __restrict__ att, float* __restrict__ avec) {
  __shared__ float sA[16 * 256];  // up to K=256 -> 16KB
  const int lane = threadIdx.x & 31;
  const int wave = threadIdx.x >> 5;
  const int row0 = blockIdx.x * 16;

  // stage the 16xK A tile via async global->LDS copies (ASYNCcnt)
  {
    const float* gbase = A + (size_t)row0 * K;
    lds_float* lbase = (lds_float*)sA;
    for (int e4 = threadIdx.x * 4; e4 < 16 * K; e4 += blockDim.x * 4)
      async_copy16(gbase + e4, lbase + e4);
    wait_async0();
  }
  __syncthreads();

  const int mrow = lane & 15;
  const int khalf = lane >> 4;
  const int nt0 = wave * 2;
  const int c0 = nt0 * 16 + mrow;
  const int c1 = c0 + 16;
  const bool v0 = (nt0 * 16) < N;
  const bool v1 = (nt0 * 16 + 16) < N;
  if (!v0) return;

  const float* sArow = sA + mrow * K + 2 * khalf;
  const float* b0p = Bt + (size_t)c0 * K + 2 * khalf;
  const float* b1p = Bt + (size_t)c1 * K + 2 * khalf;

  v8f acc0 = {}, acc1 = {};
  if (v1) {
#pragma unroll 4
    for (int k0 = 0; k0 < K; k0 += 4) {
      v2f a  = *(const v2f*)(sArow + k0);
      v2f b0 = *(const v2f*)(b0p + k0);
      v2f b1 = *(const v2f*)(b1p + k0);
      acc0 = __builtin_amdgcn_wmma_f32_16x16x4_f32(false, a, false, b0, (short)0, acc0,
                                                   false, false);
      acc1 = __builtin_amdgcn_wmma_f32_16x16x4_f32(false, a, false, b1, (short)0, acc1,
                                                   false, false);
    }
  } else {
#pragma unroll 4
    for (int k0 = 0; k0 < K; k0 += 4) {
      v2f a  = *(const v2f*)(sArow + k0);
      v2f b0 = *(const v2f*)(b0p + k0);
      acc0 = __builtin_amdgcn_wmma_f32_16x16x4_f32(false, a, false, b0, (short)0, acc0,
                                                   false, false);
    }
  }

  float* crow = C + (size_t)(row0 + 8 * khalf) * N;
#pragma unroll
  for (int v = 0; v < 8; ++v) crow[(size_t)v * N + c0] = acc0[v];
  if (v1) {
#pragma unroll
    for (int v = 0; v < 8; ++v) crow[(size_t)v * N + c1] = acc1[v];
  }

  // fused per-head attention dot (GAT GEMMs only; N == HC there)
  if (att != nullptr) {
    const int hh = (nt0 * 16) >> 6;           // head owning this wave's columns
    const float w0 = att[c0];
    const float w1 = v1 ? att[c1] : 0.0f;
#pragma unroll
    for (int v = 0; v < 8; ++v) {
      float p = acc0[v] * w0;
      if (v1) p += acc1[v] * w1;
      p += __shfl_xor(p, 1, 32);
      p += __shfl_xor(p, 2, 32);
      p += __shfl_xor(p, 4, 32);
      p += __shfl_xor(p, 8, 32);              // sum over the 16-lane group
      if (mrow == 0)
        atomicAdd(&avec[(size_t)(row0 + v + 8 * khalf) * HEADS + hh], p);
    }
  }
}

// ---------------------------------------------------------------------------
// pass A: alpha = leaky_relu(a_s[src]+a_d[dst]); segment max via atomics
__global__ void edge_alpha_kernel(const int* __restrict__ src, const int* __restrict__ dst,
                                  const float* __restrict__ a_s, const float* __restrict__ a_d,
                                  float* __restrict__ alpha, float* __restrict__ amax, int E) {
  int e = blockIdx.x * blockDim.x + threadIdx.x;
  if (e >= E) return;
  int s = src[e], d = dst[e];
#pragma unroll
  for (int h = 0; h < HEADS; ++h) {
    float al = a_s[s * HEADS + h] + a_d[d * HEADS + h];
    al = (al > 0.0f) ? al : 0.2f * al;
    alpha[(size_t)e * HEADS + h] = al;
    atomicMaxF(&amax[d * HEADS + h], al);
  }
}

// pass B: ex = exp(alpha - amax[dst]); segment sum
__global__ void edge_exp_kernel(const int* __restrict__ dst, float* __restrict__ alpha,
                                const float* __restrict__ amax, float* __restrict__ den, int E) {
  int e = blockIdx.x * blockDim.x + threadIdx.x;
  if (e >= E) return;
  int d = dst[e];
#pragma unroll
  for (int h = 0; h < HEADS; ++h) {
    float ex = __expf(alpha[(size_t)e * HEADS + h] - amax[d * HEADS + h]);
    alpha[(size_t)e * HEADS + h] = ex;
    atomicAdd(&den[d * HEADS + h], ex);
  }
}

// pass C: agg[dst] += hs[src] * (ex/den[dst]); one wave32 per edge slot,
// grid-stride over edges with prefetch of the next gather row (global_prefetch).
__global__ void edge_scatter_kernel(const int* __restrict__ src, const int* __restrict__ dst,
                                    const float* __restrict__ hs, const float* __restrict__ ex,
                                    const float* __restrict__ den, float* __restrict__ agg, int E) {
  const int lane = threadIdx.x & 31;
  const int widx = (blockIdx.x * blockDim.x + threadIdx.x) >> 5;
  const int nw = (gridDim.x * blockDim.x) >> 5;
  const int h = lane >> 3;  // channels [lane*8, lane*8+8) lie inside head lane/8

  for (int e = widx; e < E; e += nw) {
    int s = src[e], d = dst[e];
    int en = e + nw;
    if (en < E) {  // prefetch next iteration's gathered row into caches
      int sn = src[en];
      __builtin_prefetch(hs + (size_t)sn * HC + lane * 8, 0, 0);
    }
    float w = ex[(size_t)e * HEADS + h] / den[d * HEADS + h];
    const float4* hrow = (const float4*)(hs + (size_t)s * HC);
    float4 q0 = hrow[lane * 2 + 0];
    float4 q1 = hrow[lane * 2 + 1];
    float* arow = agg + (size_t)d * HC + lane * 8;
    atomicAdd(arow + 0, q0.x * w); atomicAdd(arow + 1, q0.y * w);
    atomicAdd(arow + 2, q0.z * w); atomicAdd(arow + 3, q0.w * w);
    atomicAdd(arow + 4, q1.x * w); atomicAdd(arow + 5, q1.y * w);
    atomicAdd(arow + 6, q1.z * w); atomicAdd(arow + 7, q1.w * w);
  }
}

// inter = relu(agg + bias_po + bias_adj)
__global__ void inter_kernel(const float* __restrict__ agg, const float* __restrict__ b0,
                             const float* __restrict__ b1, float* __restrict__ inter) {
  int t = blockIdx.x * blockDim.x + threadIdx.x;
  if (t >= NINT * HC) return;
  int c = t & (HC - 1);
  inter[t] = fmaxf(agg[t] + b0[c] + b1[c], 0.0f);
}

// GRU gates; writes h_new
__global__ void gru_kernel(const float* __restrict__ gi, const float* __restrict__ gh,
                           const float* __restrict__ b_ih, const float* __restrict__ b_hh,
                           const float* __restrict__ h0, float* __restrict__ hnew) {
  int t = blockIdx.x * blockDim.x + threadIdx.x;
  if (t >= NINT * CH) return;
  int n = t >> 6, c = t & 63;
  const float* gin = gi + (size_t)n * G3;
  const float* ghn = gh + (size_t)n * G3;
  float r = sigm_(gin[c]        + b_ih[c]        + ghn[c]        + b_hh[c]);
  float z = sigm_(gin[64 + c]   + b_ih[64 + c]   + ghn[64 + c]   + b_hh[64 + c]);
  float nn = tanh_(gin[128 + c] + b_ih[128 + c] + r * (ghn[128 + c] + b_hh[128 + c]));
  hnew[t] = (1.0f - z) * nn + z * h0[t];
}

// logits (8) + value (1) per node
__global__ void heads_kernel(const float* __restrict__ hnew, const float* __restrict__ Wpi,
                             const float* __restrict__ bpi, const float* __restrict__ Wv,
                             const float* __restrict__ bv, float* __restrict__ logits,
                             float* __restrict__ value) {
  int n = blockIdx.x * blockDim.x + threadIdx.x;
  if (n >= NINT) return;
  const float* h = hnew + (size_t)n * CH;
  float acc[8];
#pragma unroll
  for (int j = 0; j < 8; ++j) acc[j] = bpi[j];
  float av = bv[0];
  for (int c = 0; c < CH; ++c) {
    float hv = h[c];
#pragma unroll
    for (int j = 0; j < 8; ++j) acc[j] += hv * Wpi[c * 8 + j];
    av += hv * Wv[c];
  }
#pragma unroll
  for (int j = 0; j < 8; ++j) logits[(size_t)n * 8 + j] = acc[j];
  value[n] = av;
}

// ---------------------------------------------------------------------------
extern "C" void kernel_launch(void* const* d_in, const int* in_sizes, int n_in,
                              void* d_out, int out_size, void* d_ws, size_t ws_size,
                              hipStream_t stream) {
  const float* x_int      = (const float*)d_in[0];
  const float* x_lane     = (const float*)d_in[1];
  const float* h0         = (const float*)d_in[2];
  const float* enc_int_W  = (const float*)d_in[3];
  const float* enc_int_b  = (const float*)d_in[4];
  const float* enc_lane_W = (const float*)d_in[5];
  const float* enc_lane_b = (const float*)d_in[6];
  const float* Wsrc_po    = (const float*)d_in[7];
  const float* Wdst_po    = (const float*)d_in[8];
  const float* att_src_po = (const float*)d_in[9];
  const float* att_dst_po = (const float*)d_in[10];
  const float* bias_po    = (const float*)d_in[11];
  const float* Wsrc_adj   = (const float*)d_in[12];
  const float* Wdst_adj   = (const float*)d_in[13];
  const float* att_src_adj= (const float*)d_in[14];
  const float* att_dst_adj= (const float*)d_in[15];
  const float* bias_adj   = (const float*)d_in[16];
  const float* gru_W_ih   = (const float*)d_in[17];
  const float* gru_W_hh   = (const float*)d_in[18];
  const float* gru_b_ih   = (const float*)d_in[19];
  const float* gru_b_hh   = (const float*)d_in[20];
  const float* W_pi       = (const float*)d_in[21];
  const float* b_pi       = (const float*)d_in[22];
  const float* W_v        = (const float*)d_in[23];
  const float* b_v        = (const float*)d_in[24];
  const int*   src_po     = (const int*)d_in[25];
  const int*   dst_po     = (const int*)d_in[26];
  const int*   src_adj    = (const int*)d_in[27];
  const int*   dst_adj    = (const int*)d_in[28];

  // workspace layout (floats)
  float* ws = (float*)d_ws;
  size_t o = 0;
  float* e_lane   = ws + o; o += (size_t)NLANE * CH;   // encoded lanes
  float* e_int    = ws + o; o += (size_t)NINT * CH;    // encoded intersections
  float* hs_po    = ws + o; o += (size_t)NLANE * HC;   // lane @ Wsrc_po
  float* hd_po    = ws + o; o += (size_t)NINT * HC;
  float* hs_adj   = ws + o; o += (size_t)NINT * HC;
  float* hd_adj   = ws + o; o += (size_t)NINT * HC;
  float* as_po    = ws + o; o += (size_t)NLANE * HEADS;  // -- atomically accumulated
  float* ad_po    = ws + o; o += (size_t)NINT * HEADS;   //    (contiguous, one memset)
  float* as_adj   = ws + o; o += (size_t)NINT * HEADS;
  float* ad_adj   = ws + o; o += (size_t)NINT * HEADS;
  float* alpha_po = ws + o; o += (size_t)EPO * HEADS;
  float* alpha_adj= ws + o; o += (size_t)EADJ * HEADS;
  float* amax_po  = ws + o; o += (size_t)NINT * HEADS;
  float* amax_adj = ws + o; o += (size_t)NINT * HEADS;
  float* den_po   = ws + o; o += (size_t)NINT * HEADS;
  float* den_adj  = ws + o; o += (size_t)NINT * HEADS;
  float* agg      = ws + o; o += (size_t)NINT * HC;    // shared sum of both relations
  float* inter    = ws + o; o += (size_t)NINT * HC;
  float* gi       = ws + o; o += (size_t)NINT * G3;
  float* gh       = ws + o; o += (size_t)NINT * G3;
  float* WsrcT_po = ws + o; o += (size_t)HC * CH;      // transposed GAT weights [256][64]
  float* WdstT_po = ws + o; o += (size_t)HC * CH;
  float* WsrcT_adj= ws + o; o += (size_t)HC * CH;
  float* WdstT_adj= ws + o; o += (size_t)HC * CH;

  float* out_logits = (float*)d_out;                 // [NINT,8]
  float* out_value  = out_logits + (size_t)NINT * 8; // [NINT]
  float* out_hnew   = out_value + NINT;              // [NINT,CH]

  const int B = 256;

  // init accumulators (graph-capture-safe async ops only)
  (void)hipMemsetAsync(den_po, 0, (size_t)2 * NINT * HEADS * sizeof(float), stream);
  (void)hipMemsetAsync(agg, 0, (size_t)NINT * HC * sizeof(float), stream);
  (void)hipMemsetAsync(as_po, 0, (size_t)(NLANE + 3 * NINT) * HEADS * sizeof(float), stream);
  fill_kernel<<<(2 * NINT * HEADS + B - 1) / B, B, 0, stream>>>(
      amax_po, 2 * NINT * HEADS, -__builtin_inff());

  // transpose GAT weights to [N][K] so all GEMMs are branch-free TN form
  const int WSZ = CH * HC;  // 16384
  transpose_kernel<<<(WSZ + B - 1) / B, B, 0, stream>>>(Wsrc_po, WsrcT_po, CH, HC);
  transpose_kernel<<<(WSZ + B - 1) / B, B, 0, stream>>>(Wdst_po, WdstT_po, CH, HC);
  transpose_kernel<<<(WSZ + B - 1) / B, B, 0, stream>>>(Wsrc_adj, WsrcT_adj, CH, HC);
  transpose_kernel<<<(WSZ + B - 1) / B, B, 0, stream>>>(Wdst_adj, WdstT_adj, CH, HC);

  // encoders
  enc_kernel<<<(NLANE * CH + B - 1) / B, B, 0, stream>>>(x_lane, enc_lane_W, enc_lane_b,
                                                         e_lane, NLANE, 12);
  enc_kernel<<<(NINT * CH + B - 1) / B, B, 0, stream>>>(x_int, enc_int_W, enc_int_b,
                                                        e_int, NINT, 16);

  // GAT projections via fp32 WMMA with fused per-head attention dots
  wmma_gemm_tn<<<NLANE / 16, B, 0, stream>>>(e_lane, WsrcT_po, hs_po, NLANE, HC, CH,
                                             att_src_po, as_po);
  wmma_gemm_tn<<<NINT / 16, B, 0, stream>>>(e_int, WdstT_po, hd_po, NINT, HC, CH,
                                            att_dst_po, ad_po);
  wmma_gemm_tn<<<NINT / 16, B, 0, stream>>>(e_int, WsrcT_adj, hs_adj, NINT, HC, CH,
                                            att_src_adj, as_adj);
  wmma_gemm_tn<<<NINT / 16, B, 0, stream>>>(e_int, WdstT_adj, hd_adj, NINT, HC, CH,
                                            att_dst_adj, ad_adj);

  // edge softmax + scatter, relation po (lane -> intersection)
  edge_alpha_kernel<<<(EPO + B - 1) / B, B, 0, stream>>>(src_po, dst_po, as_po, ad_po,
                                                         alpha_po, amax_po, EPO);
  edge_exp_kernel<<<(EPO + B - 1) / B, B, 0, stream>>>(dst_po, alpha_po, amax_po, den_po, EPO);
  edge_scatter_kernel<<<(EPO + 31) / 32, B, 0, stream>>>(src_po, dst_po, hs_po, alpha_po,
                                                         den_po, agg, EPO);
  // relation adj (intersection -> intersection), summed into same agg
  edge_alpha_kernel<<<(EADJ + B - 1) / B, B, 0, stream>>>(src_adj, dst_adj, as_adj, ad_adj,
                                                          alpha_adj, amax_adj, EADJ);
  edge_exp_kernel<<<(EADJ + B - 1) / B, B, 0, stream>>>(dst_adj, alpha_adj, amax_adj, den_adj, EADJ);
  edge_scatter_kernel<<<(EADJ + 31) / 32, B, 0, stream>>>(src_adj, dst_adj, hs_adj, alpha_adj,
                                                          den_adj, agg, EADJ);

  // inter = relu(agg + biases)
  inter_kernel<<<(NINT * HC + B - 1) / B, B, 0, stream>>>(agg, bias_po, bias_adj, inter);

  // GRU GEMMs (weights already [N][K]): gi = inter @ W_ih^T, gh = h0 @ W_hh^T
  wmma_gemm_tn<<<NINT / 16, B, 0, stream>>>(inter, gru_W_ih, gi, NINT, G3, HC, nullptr, nullptr);
  wmma_gemm_tn<<<NINT / 16, B, 0, stream>>>(h0, gru_W_hh, gh, NINT, G3, CH, nullptr, nullptr);

  // gates + outputs
  gru_kernel<<<(NINT * CH + B - 1) / B, B, 0, stream>>>(gi, gh, gru_b_ih, gru_b_hh, h0, out_hnew);
  heads_kernel<<<(NINT + B - 1) / B, B, 0, stream>>>(out_hnew, W_pi, b_pi, W_v, b_v,
                                                     out_logits, out_value);
}